// DiffJPEG_11269994185510
// MI455X (gfx1250) — compile-verified
//
#include <hip/hip_runtime.h>
#include <math.h>

// CDNA5 / gfx1250: wave32, WMMA f32 16x16x4.
typedef __attribute__((ext_vector_type(2))) float v2f;
typedef __attribute__((ext_vector_type(8))) float v8f;

#define QUALITY_FACTOR 0.5f   // quality=75 -> (200-150)/100
#define QF(v) ((v) * QUALITY_FACTOR)

// Combined constant table, staged to LDS at kernel start:
//  [0..63]   D8[u][x] = 0.5*cos((2x+1)*u*pi/16), row 0 scaled by 1/sqrt(2)
//  [64..127] Y quant table * factor
//  [128..191] C quant table * factor
__device__ __constant__ __align__(16) float cTab[192] = {
  // ---- D8 ----
  0.3535533906f, 0.3535533906f, 0.3535533906f, 0.3535533906f,
  0.3535533906f, 0.3535533906f, 0.3535533906f, 0.3535533906f,
  0.4903926402f, 0.4157348062f, 0.2777851165f, 0.0975451610f,
 -0.0975451610f,-0.2777851165f,-0.4157348062f,-0.4903926402f,
  0.4619397663f, 0.1913417162f,-0.1913417162f,-0.4619397663f,
 -0.4619397663f,-0.1913417162f, 0.1913417162f, 0.4619397663f,
  0.4157348062f,-0.0975451610f,-0.4903926402f,-0.2777851165f,
  0.2777851165f, 0.4903926402f, 0.0975451610f,-0.4157348062f,
  0.3535533906f,-0.3535533906f,-0.3535533906f, 0.3535533906f,
  0.3535533906f,-0.3535533906f,-0.3535533906f, 0.3535533906f,
  0.2777851165f,-0.4903926402f, 0.0975451610f, 0.4157348062f,
 -0.4157348062f,-0.0975451610f, 0.4903926402f,-0.2777851165f,
  0.1913417162f,-0.4619397663f, 0.4619397663f,-0.1913417162f,
 -0.1913417162f, 0.4619397663f,-0.4619397663f, 0.1913417162f,
  0.0975451610f,-0.2777851165f, 0.4157348062f,-0.4903926402f,
  0.4903926402f,-0.4157348062f, 0.2777851165f,-0.0975451610f,
  // ---- Y quant * factor ----
 QF(16),QF(11),QF(10),QF(16),QF(24),QF(40),QF(51),QF(61),
 QF(12),QF(12),QF(14),QF(19),QF(26),QF(58),QF(60),QF(55),
 QF(14),QF(13),QF(16),QF(24),QF(40),QF(57),QF(69),QF(56),
 QF(14),QF(17),QF(22),QF(29),QF(51),QF(87),QF(80),QF(62),
 QF(18),QF(22),QF(37),QF(56),QF(68),QF(109),QF(103),QF(77),
 QF(24),QF(35),QF(55),QF(64),QF(81),QF(104),QF(113),QF(92),
 QF(49),QF(64),QF(78),QF(87),QF(103),QF(121),QF(120),QF(101),
 QF(72),QF(92),QF(95),QF(98),QF(112),QF(100),QF(103),QF(99),
  // ---- C quant * factor ----
 QF(17),QF(18),QF(24),QF(47),QF(99),QF(99),QF(99),QF(99),
 QF(18),QF(21),QF(26),QF(66),QF(99),QF(99),QF(99),QF(99),
 QF(24),QF(26),QF(56),QF(99),QF(99),QF(99),QF(99),QF(99),
 QF(47),QF(66),QF(99),QF(99),QF(99),QF(99),QF(99),QF(99),
 QF(99),QF(99),QF(99),QF(99),QF(99),QF(99),QF(99),QF(99),
 QF(99),QF(99),QF(99),QF(99),QF(99),QF(99),QF(99),QF(99),
 QF(99),QF(99),QF(99),QF(99),QF(99),QF(99),QF(99),QF(99),
 QF(99),QF(99),QF(99),QF(99),QF(99),QF(99),QF(99),QF(99)};

// One multiply stage: D = A(16x16) * X^T(16x16), K accumulated as 4 chunks of 4.
// X is a 16x16 row-major LDS tile; B-layout of X^T means lane n reads row n of X.
__device__ inline v8f wstage(const v2f A[4], const float* __restrict__ X, int n, int half) {
  v8f acc = {};
#pragma unroll
  for (int c = 0; c < 4; ++c) {
    const int col0 = 4 * c + 2 * half;                      // K pair held by this lane-half
    const float2 t = *(const float2*)(X + n * 16 + col0);   // ds_load_b64
    v2f bv; bv.x = t.x; bv.y = t.y;
    acc = __builtin_amdgcn_wmma_f32_16x16x4_f32(
        /*neg_a=*/false, A[c], /*neg_b=*/false, bv,
        /*c_mod=*/(short)0, acc, /*reuse_a=*/false, /*reuse_b=*/false);
  }
  return acc;
}

// Store a D-layout 16x16 f32 accumulator (8 VGPRs) to row-major LDS.
__device__ inline void dstore(float* __restrict__ X, v8f d, int n, int half) {
#pragma unroll
  for (int v = 0; v < 8; ++v) X[(v + 8 * half) * 16 + n] = d[v];
}

__global__ __launch_bounds__(32)
void diffjpeg_kernel(const float* __restrict__ x, float* __restrict__ out) {
  __shared__ __align__(16) float s0[256];   // Y tile / quantized coef / rec Y
  __shared__ __align__(16) float s1[256];   // chroma tile / coef / rec
  __shared__ __align__(16) float s2[256];   // stage temp Y
  __shared__ __align__(16) float s3[256];   // stage temp chroma
  __shared__ __align__(16) float sT[192];   // staged tables: D8 | Qy | Qc

  const int tid  = threadIdx.x;
  const int n    = tid & 15;      // D-layout column / B-layout source row
  const int half = tid >> 4;

  const int gid = blockIdx.x;
  const int b   = gid >> 10;      // 1024 tiles per 512x512 image
  const int t   = gid & 1023;
  const int ty  = t >> 5, tx = t & 31;

  // ---------------- stage constant tables into LDS (one b128 clause) ----------
  {
    float4 t0 = *(const float4*)&cTab[tid * 4];          // covers [0..127]
    *(float4*)&sT[tid * 4] = t0;
    if (tid < 16) {
      float4 t1 = *(const float4*)&cTab[128 + tid * 4];  // covers [128..191]
      *(float4*)&sT[128 + tid * 4] = t1;
    }
  }

  // ---------------- load 16x16x3 pixels, color convert ----------------
  const int row  = tid >> 1;            // 0..15
  const int cb8  = (tid & 1) * 8;       // 0 or 8
  const size_t plane  = 512 * 512;
  const size_t pixoff = (size_t)(ty * 16 + row) * 512 + (tx * 16 + cb8);

  float R[8], G[8], Bl[8];
  {
    const float* pr = x + ((size_t)b * 3 + 0) * plane + pixoff;
    const float* pg = x + ((size_t)b * 3 + 1) * plane + pixoff;
    const float* pb = x + ((size_t)b * 3 + 2) * plane + pixoff;
    *(float4*)&R[0]  = ((const float4*)pr)[0];  *(float4*)&R[4]  = ((const float4*)pr)[1];
    *(float4*)&G[0]  = ((const float4*)pg)[0];  *(float4*)&G[4]  = ((const float4*)pg)[1];
    *(float4*)&Bl[0] = ((const float4*)pb)[0];  *(float4*)&Bl[4] = ((const float4*)pb)[1];
  }
  __attribute__((aligned(16))) float ya[8];
  float cbv[8], crv[8];
#pragma unroll
  for (int j = 0; j < 8; ++j) {
    const float r = fminf(fmaxf(R[j],  0.0f), 1.0f) * 255.0f;
    const float g = fminf(fmaxf(G[j],  0.0f), 1.0f) * 255.0f;
    const float bb= fminf(fmaxf(Bl[j], 0.0f), 1.0f) * 255.0f;
    ya[j]  = 0.299f * r + 0.587f * g + 0.114f * bb - 128.0f;   // y - 128
    cbv[j] = -0.168736f * r - 0.331264f * g + 0.5f * bb;       // cb - 128
    crv[j] =  0.5f * r - 0.418688f * g - 0.081312f * bb;       // cr - 128
  }
  *(float4*)&s0[row * 16 + cb8]     = *(float4*)&ya[0];
  *(float4*)&s0[row * 16 + cb8 + 4] = *(float4*)&ya[4];

  // ---------------- 2x2 chroma subsample via shuffles ----------------
  // lane l holds full-res row l>>1, cols (l&1)*8..+7; vertical partner = lane^2.
  __attribute__((aligned(16))) float hb[4], hr[4];
#pragma unroll
  for (int j = 0; j < 4; ++j) {
    hb[j] = cbv[2 * j] + cbv[2 * j + 1];
    hr[j] = crv[2 * j] + crv[2 * j + 1];
  }
#pragma unroll
  for (int j = 0; j < 4; ++j) {
    hb[j] = 0.25f * (hb[j] + __shfl_xor(hb[j], 2, 32));
    hr[j] = 0.25f * (hr[j] + __shfl_xor(hr[j], 2, 32));
  }
  {  // bottom half (rows 8..15) of chroma tile = 0: 4 entries per lane
    const float4 z4 = make_float4(0.0f, 0.0f, 0.0f, 0.0f);
    *(float4*)&s1[128 + tid * 4] = z4;
  }
  if ((tid & 2) == 0) {           // lanes holding even full-res rows own sub-row tid>>2
    const int i  = tid >> 2;      // 0..7
    const int c0 = (tid & 1) * 4; // 0 or 4
    *(float4*)&s1[i * 16 + c0]     = *(float4*)&hb[0];   // Cb quadrant (0,0)
    *(float4*)&s1[i * 16 + 8 + c0] = *(float4*)&hr[0];   // Cr quadrant (0,1)
  }
  __syncthreads();

  // ---------------- constant A operands: Dd (fwd) and Dd^T (inv) ----------------
  v2f Af[4], Ai[4];
#pragma unroll
  for (int c = 0; c < 4; ++c) {
    const int k0  = 4 * c + 2 * half;
    const bool on = (n < 8) == (k0 < 8);        // block-diagonal mask
    const float2 f = *(const float2*)&sT[(n & 7) * 8 + (k0 & 7)];   // ds_load_b64
    Af[c].x = on ? f.x : 0.0f;
    Af[c].y = on ? f.y : 0.0f;
    const float g0 = sT[((k0)     & 7) * 8 + (n & 7)];  // pair stride 8 ->
    const float g1 = sT[((k0 + 1) & 7) * 8 + (n & 7)];  // ds_load_2addr_b32
    Ai[c].x = on ? g0 : 0.0f;
    Ai[c].y = on ? g1 : 0.0f;
  }

  // Stage 1: U = Dd * X^T
  v8f Uy = wstage(Af, s0, n, half);
  v8f Uc = wstage(Af, s1, n, half);
  dstore(s2, Uy, n, half);
  dstore(s3, Uc, n, half);
  __syncthreads();

  // Stage 2: C = Dd * U^T = Dd X Dd^T ; quantize
  v8f Cy = wstage(Af, s2, n, half);
  v8f Cc = wstage(Af, s3, n, half);
  {
    const float* qp = &sT[64 + (n & 7)];  // coef (m,n): q idx = (m&7)*8+(n&7); m&7 == v
#pragma unroll
    for (int v = 0; v < 8; ++v) {
      const float qy = qp[8 * v];         // ds_load, immediate offsets
      const float qc = qp[64 + 8 * v];
      Cy[v] = rintf(Cy[v] / qy) * qy;     // v_rndne = round-half-even, matches jnp.round
      Cc[v] = rintf(Cc[v] / qc) * qc;
    }
  }
  dstore(s0, Cy, n, half);
  dstore(s1, Cc, n, half);
  __syncthreads();

  // Stage 3: V = Dd^T * Cq^T
  v8f Vy = wstage(Ai, s0, n, half);
  v8f Vc = wstage(Ai, s1, n, half);
  dstore(s2, Vy, n, half);
  dstore(s3, Vc, n, half);
  __syncthreads();

  // Stage 4: R = Dd^T * V^T = Dd^T Cq Dd (inverse DCT)
  v8f Ry = wstage(Ai, s2, n, half);
  v8f Rc = wstage(Ai, s3, n, half);
  dstore(s0, Ry, n, half);   // rec_y - 128
  dstore(s1, Rc, n, half);   // rows 0..7: [cbm | crm]
  __syncthreads();

  // ---------------- upsample + YCbCr->RGB + clip, store ----------------
  float4 y0  = *(const float4*)&s0[row * 16 + cb8];
  float4 y1  = *(const float4*)&s0[row * 16 + cb8 + 4];
  float4 cb4 = *(const float4*)&s1[(row >> 1) * 16 + (cb8 >> 1)];
  float4 cr4 = *(const float4*)&s1[(row >> 1) * 16 + 8 + (cb8 >> 1)];
  const float yv8[8]  = {y0.x, y0.y, y0.z, y0.w, y1.x, y1.y, y1.z, y1.w};
  const float cbm4[4] = {cb4.x, cb4.y, cb4.z, cb4.w};
  const float crm4[4] = {cr4.x, cr4.y, cr4.z, cr4.w};

  __attribute__((aligned(16))) float oR[8], oG[8], oB[8];
#pragma unroll
  for (int j = 0; j < 8; ++j) {
    const float yv  = yv8[j] + 128.0f;
    const float cbm = cbm4[j >> 1];
    const float crm = crm4[j >> 1];
    const float r2 = yv + 1.402f * crm;
    const float g2 = yv - 0.344136f * cbm - 0.714136f * crm;
    const float b2 = yv + 1.772f * cbm;
    oR[j] = fminf(fmaxf(r2, 0.0f), 255.0f) * (1.0f / 255.0f);
    oG[j] = fminf(fmaxf(g2, 0.0f), 255.0f) * (1.0f / 255.0f);
    oB[j] = fminf(fmaxf(b2, 0.0f), 255.0f) * (1.0f / 255.0f);
  }
  {
    float* qr = out + ((size_t)b * 3 + 0) * plane + pixoff;
    float* qg = out + ((size_t)b * 3 + 1) * plane + pixoff;
    float* qb = out + ((size_t)b * 3 + 2) * plane + pixoff;
    ((float4*)qr)[0] = *(float4*)&oR[0];  ((float4*)qr)[1] = *(float4*)&oR[4];
    ((float4*)qg)[0] = *(float4*)&oG[0];  ((float4*)qg)[1] = *(float4*)&oG[4];
    ((float4*)qb)[0] = *(float4*)&oB[0];  ((float4*)qb)[1] = *(float4*)&oB[4];
  }
}

extern "C" void kernel_launch(void* const* d_in, const int* in_sizes, int n_in,
                              void* d_out, int out_size, void* d_ws, size_t ws_size,
                              hipStream_t stream) {
  (void)n_in; (void)out_size; (void)d_ws; (void)ws_size;
  const float* x = (const float*)d_in[0];
  float* out = (float*)d_out;
  const int nb = in_sizes[0] / (3 * 512 * 512);   // batch
  const int nblocks = nb * 32 * 32;               // one 16x16 tile per single-wave WG
  diffjpeg_kernel<<<nblocks, 32, 0, stream>>>(x, out);
}